// ChebConv_1580547967739
// MI455X (gfx1250) — compile-verified
//
#include <hip/hip_runtime.h>

// Problem constants (from reference)
#define V_N   50000
#define E_N   800000
#define B_N   2
#define FIN   64
#define FOUT  64
#define KCH   4
#define XCOLS (FIN * B_N)   // 128 columns of each Chebyshev state
#define MROWS (B_N * V_N)   // 100000 GEMM rows
#define KDIM  (KCH * FIN)   // 256 GEMM reduction dim
#define XPITCH 260          // sX row pitch (+4 floats: kills 16-way bank conflict)

typedef __attribute__((ext_vector_type(2))) float v2f;
typedef __attribute__((ext_vector_type(8))) float v8f;

// ---------------------------------------------------------------------------
// x0[v, f*B + b] = inputs[b, v, f]
// ---------------------------------------------------------------------------
__global__ __launch_bounds__(256) void pack_x0(const float* __restrict__ inp,
                                               float* __restrict__ x0) {
  int i = blockIdx.x * 256 + threadIdx.x;
  if (i >= V_N * XCOLS) return;
  int b = i & 1;
  int f = (i >> 1) & 63;
  int v = i >> 7;
  x0[i] = inp[((size_t)b * V_N + v) * FIN + f];
}

__global__ __launch_bounds__(256) void set_zero(float* __restrict__ p, int n) {
  int i = blockIdx.x * 256 + threadIdx.x;
  if (i < n) p[i] = 0.0f;
}

// dst = -src  (Chebyshev recurrence:  x_k = 2*L*x_{k-1} - x_{k-2})
__global__ __launch_bounds__(256) void set_neg(float* __restrict__ dst,
                                               const float* __restrict__ src, int n) {
  int i = blockIdx.x * 256 + threadIdx.x;
  if (i < n) dst[i] = -src[i];
}

// ---------------------------------------------------------------------------
// y[row] += scale * vals[e] * x[col]   (COO SpMM, 32 lanes per edge,
// float4 gather + 4 global_atomic_add_f32 per lane; data is L2-resident)
// ---------------------------------------------------------------------------
__global__ __launch_bounds__(256) void spmm_atomic(const float* __restrict__ x,
                                                   float* __restrict__ y,
                                                   const float* __restrict__ vals,
                                                   const int* __restrict__ rows,
                                                   const int* __restrict__ cols,
                                                   float scale) {
  int t = blockIdx.x * 256 + threadIdx.x;
  int e = t >> 5;
  int lane = t & 31;
  if (e >= E_N) return;
  int r = rows[e];
  int c = cols[e];
  float a = scale * vals[e];
  float4 xv = ((const float4*)(x + (size_t)c * XCOLS))[lane];
  float* dst = y + (size_t)r * XCOLS + lane * 4;
  atomicAdd(dst + 0, a * xv.x);
  atomicAdd(dst + 1, a * xv.y);
  atomicAdd(dst + 2, a * xv.z);
  atomicAdd(dst + 3, a * xv.w);
}

// ---------------------------------------------------------------------------
// out[m, n] = sum_c X[m, c] * Wflat[c, n] + bias[n]
//   m = b*V + v ; X[m, c] = xk[c&3][ v*128 + (c>>2)*2 + b ]
//   Wflat = weight contiguous as [256, 64] (faithful to reference flatten)
//
// 4 waves / block, 64 M-rows / block, full N=64 per wave as 4 16x16 tiles,
// K=256 via V_WMMA_F32_16X16X4_F32 (full fp32 precision).
//
// Weights staged in LDS pre-swizzled into per-lane B fragments:
//   sWs[ks*128 + n*2 + h] = { Wf[ks*4+2h][n], Wf[ks*4+2h+1][n] }
// so each B fragment is one 8B ds_load and the 32 lanes of a wave stripe
// across all 64 LDS banks.
// ---------------------------------------------------------------------------
__global__ __launch_bounds__(128) void cheb_gemm_wmma(
    const float* __restrict__ xb0, const float* __restrict__ xb1,
    const float* __restrict__ xb2, const float* __restrict__ xb3,
    const float* __restrict__ weight, const float* __restrict__ bias,
    float* __restrict__ out) {
  __shared__ v2f   sWs[(KDIM / 2) * FOUT];  // 8192 pairs = 64 KB
  __shared__ float sX[64 * XPITCH];         // 65 KB (padded pitch)

  int tid = threadIdx.x;
  int m_block = blockIdx.x * 64;

  // Stage weights, swizzled into B-fragment layout.
#pragma unroll
  for (int i = 0; i < 64; ++i) {
    int j = i * 128 + tid;      // 0..8191
    int ks  = j >> 7;           // k-step 0..63
    int rem = j & 127;
    int n = rem >> 1;           // 0..63
    int h = rem & 1;            // half
    v2f w;
    w.x = weight[(ks * 4 + 2 * h) * FOUT + n];
    w.y = weight[(ks * 4 + 2 * h + 1) * FOUT + n];
    sWs[j] = w;
  }

  // Stage X tile: gather from the 4 Chebyshev state buffers with the
  // reference's (intentional) column pairing c = f*K + k.
  for (int i = 0; i < 128; ++i) {
    int elem = i * 128 + tid;
    int r = elem >> 8;    // 0..63
    int c = elem & 255;   // 0..255
    int m = m_block + r;
    float val = 0.0f;
    if (m < MROWS) {
      int b = (m >= V_N) ? 1 : 0;
      int v = m - b * V_N;
      int k = c & 3;
      int f = c >> 2;
      const float* xk = (k == 0) ? xb0 : (k == 1) ? xb1 : (k == 2) ? xb2 : xb3;
      val = xk[(size_t)v * XCOLS + f * B_N + b];
    }
    sX[r * XPITCH + c] = val;
  }
  __syncthreads();

  int wave = tid >> 5;
  int lane = tid & 31;
  int half = lane >> 4;   // 0: lanes 0-15, 1: lanes 16-31
  int mi   = lane & 15;
  int koff = 2 * half;    // A K-slot per ISA 16x4 f32 layout

  const float* ax = &sX[(wave * 16 + mi) * XPITCH];
  const v2f*   bw = sWs + mi * 2 + half;

  v8f acc0 = {}, acc1 = {}, acc2 = {}, acc3 = {};
#pragma unroll 4
  for (int ks = 0; ks < 64; ++ks) {
    // A: lane(half,mi) holds A[mi][4ks+koff], A[mi][4ks+koff+1] (8B aligned)
    v2f a = *(const v2f*)&ax[ks * 4 + koff];
    // B: one b64 fragment per n-tile, pre-paired in LDS
    v2f b0 = bw[ks * 128 +  0];
    v2f b1 = bw[ks * 128 + 32];
    v2f b2 = bw[ks * 128 + 64];
    v2f b3 = bw[ks * 128 + 96];
    acc0 = __builtin_amdgcn_wmma_f32_16x16x4_f32(false, a, false, b0, (short)0, acc0, false, false);
    acc1 = __builtin_amdgcn_wmma_f32_16x16x4_f32(false, a, false, b1, (short)0, acc1, false, false);
    acc2 = __builtin_amdgcn_wmma_f32_16x16x4_f32(false, a, false, b2, (short)0, acc2, false, false);
    acc3 = __builtin_amdgcn_wmma_f32_16x16x4_f32(false, a, false, b3, (short)0, acc3, false, false);
  }

  // C/D layout: vgpr r -> (M = r + 8*half, N = mi) within the 16x16 tile
  int m_wave = m_block + wave * 16 + 8 * half;
  float bn0 = bias[ 0 + mi];
  float bn1 = bias[16 + mi];
  float bn2 = bias[32 + mi];
  float bn3 = bias[48 + mi];
#pragma unroll
  for (int r = 0; r < 8; ++r) {
    int m = m_wave + r;
    if (m < MROWS) {
      float* o = out + (size_t)m * FOUT + mi;
      o[ 0] = acc0[r] + bn0;
      o[16] = acc1[r] + bn1;
      o[32] = acc2[r] + bn2;
      o[48] = acc3[r] + bn3;
    }
  }
}

// ---------------------------------------------------------------------------
extern "C" void kernel_launch(void* const* d_in, const int* in_sizes, int n_in,
                              void* d_out, int out_size, void* d_ws, size_t ws_size,
                              hipStream_t stream) {
  const float* inputs = (const float*)d_in[0];
  const float* weight = (const float*)d_in[1];
  const float* bias   = (const float*)d_in[2];
  const float* vals   = (const float*)d_in[3];
  const int*   rows   = (const int*)d_in[4];
  const int*   cols   = (const int*)d_in[5];
  float* out = (float*)d_out;

  size_t nbuf = (size_t)V_N * XCOLS;            // 6.4M floats per state
  float* xb0 = (float*)d_ws;
  float* xb1 = xb0 + nbuf;
  float* xb2 = xb1 + nbuf;
  float* xb3 = xb2 + nbuf;

  int n = (int)nbuf;
  int nb = (n + 255) / 256;
  int spmm_blocks = (E_N * 32 + 255) / 256;

  // x0 = packed inputs
  pack_x0<<<nb, 256, 0, stream>>>(inputs, xb0);
  // x1 = L x0
  set_zero<<<nb, 256, 0, stream>>>(xb1, n);
  spmm_atomic<<<spmm_blocks, 256, 0, stream>>>(xb0, xb1, vals, rows, cols, 1.0f);
  // x2 = 2 L x1 - x0
  set_neg<<<nb, 256, 0, stream>>>(xb2, xb0, n);
  spmm_atomic<<<spmm_blocks, 256, 0, stream>>>(xb1, xb2, vals, rows, cols, 2.0f);
  // x3 = 2 L x2 - x1
  set_neg<<<nb, 256, 0, stream>>>(xb3, xb1, n);
  spmm_atomic<<<spmm_blocks, 256, 0, stream>>>(xb2, xb3, vals, rows, cols, 2.0f);
  // out = X @ Wflat + bias  (fp32 WMMA)
  cheb_gemm_wmma<<<(MROWS + 63) / 64, 128, 0, stream>>>(xb0, xb1, xb2, xb3,
                                                        weight, bias, out);
}